// GlobalLocalPool_14310831030572
// MI455X (gfx1250) — compile-verified
//
#include <hip/hip_runtime.h>
#include <stdint.h>

// Problem constants (from the reference)
constexpr int B = 16;
constexpr int T = 4096;
constexpr int H = 512;

constexpr int TILE_H   = 16;            // WMMA N dimension
constexpr int NTILES_H = H / TILE_H;    // 32 h-tiles
constexpr int WAVES    = 16;            // waves per block
constexpr int BLOCK    = WAVES * 32;    // 512 threads
constexpr int CHUNK_T  = T / WAVES;     // 256 tokens per wave
constexpr int ITERS    = CHUNK_T / 4;   // 64 WMMA (K=4) steps per wave

typedef __attribute__((ext_vector_type(2))) float v2f;
typedef __attribute__((ext_vector_type(8))) float v8f;

// One block computes out[b, hbase:hbase+16] (global view) and
// out[b, H+hbase:H+hbase+16] (local/span view) via V_WMMA_F32_16X16X4_F32:
//   D = A(16x4 masks) x B(4x16 x-tile) + C
// A rows: row 0 = validity mask (t < lengths[b]), row 1 = span mask, rest 0.
// The mask streams are pre-expanded to f32 in LDS so the hot loop is only:
//   ds_load_b64 (A) + 2x global_load_b32 (B) + v_wmma + pointer bumps.
__global__ __launch_bounds__(BLOCK)
void GlobalLocalPool_kernel(const float* __restrict__ x,
                            const int* __restrict__ lengths,
                            const unsigned char* __restrict__ mask,
                            float* __restrict__ out) {
    const int bid   = blockIdx.x;
    const int b     = bid / NTILES_H;
    const int ht    = bid % NTILES_H;
    const int hbase = ht * TILE_H;

    const int tid  = threadIdx.x;
    const int wid  = tid >> 5;         // wave id within block (0..15)
    const int lane = tid & 31;         // lane within wave
    const int m    = lane & 15;        // A-matrix row / B-matrix column (N)
    const int kh   = lane >> 4;        // 0 -> K=0,1 ; 1 -> K=2,3

    const int glen = lengths[b];
    const unsigned char* mask_row = mask + (size_t)b * T;
    const float* xb = x + (size_t)b * T * H;

    __shared__ __align__(16) float s_valid[T];     // 16 KB
    __shared__ __align__(16) float s_span[T];      // 16 KB
    __shared__ __align__(16) float s_zero[4];
    __shared__ float part[WAVES][2][TILE_H];
    __shared__ int s_slen;

    if (tid == 0) {
        s_slen = 0;
        s_zero[0] = 0.0f; s_zero[1] = 0.0f; s_zero[2] = 0.0f; s_zero[3] = 0.0f;
    }
    __syncthreads();

    // ---- expand masks to f32 streams in LDS; count span length (int atomics
    //      = order-independent = deterministic). 8 tokens per thread.
    {
        const int base = tid * (T / BLOCK);                 // 8-byte aligned
        const uint2 mv = *(const uint2*)(mask_row + base);  // 8 mask bytes
        int cnt = 0;
        #pragma unroll
        for (int j = 0; j < 8; ++j) {
            const uint32_t byte = ((j < 4 ? mv.x : mv.y) >> ((j & 3) * 8)) & 0xFFu;
            const float sv = byte ? 1.0f : 0.0f;
            cnt += byte ? 1 : 0;
            s_span[base + j]  = sv;
            s_valid[base + j] = (base + j < glen) ? 1.0f : 0.0f;
        }
        atomicAdd(&s_slen, cnt);
    }
    __syncthreads();

    // ---- per-lane A stream pointer: lanes m==0 walk s_valid, m==1 walk
    //      s_span, all others repeatedly read the zero buffer (stride 0).
    const int tbase = wid * CHUNK_T;
    const float* abase;
    int astride;
    if (m == 0)      { abase = s_valid + tbase + kh * 2; astride = 4; }
    else if (m == 1) { abase = s_span  + tbase + kh * 2; astride = 4; }
    else             { abase = s_zero;                   astride = 0; }

    // B pointer: lane m covers x[t, hbase+m] for t = tbase+kh*2 (+1 at +H)
    const float* p = xb + (size_t)(tbase + kh * 2) * H + hbase + m;

    v8f c = {};
    #pragma unroll 8
    for (int it = 0; it < ITERS; ++it) {
        const v2f a = *(const v2f*)abase;   // ds_load_b64 (8B aligned)
        abase += astride;
        v2f bv;
        bv.x = p[0];                        // global_load_b32
        bv.y = p[H];                        // global_load_b32 offset 2048
        p += 4 * H;                         // +0x2000 bytes
        c = __builtin_amdgcn_wmma_f32_16x16x4_f32(
                /*neg_a=*/false, a, /*neg_b=*/false, bv,
                /*c_mod=*/(short)0, c, /*reuse_a=*/false, /*reuse_b=*/false);
    }

    // c[0] lanes 0-15: M=0 (global sums, N=lane); c[1] lanes 0-15: M=1 (span)
    if (lane < 16) {
        part[wid][0][lane] = c[0];
        part[wid][1][lane] = c[1];
    }
    __syncthreads();

    // ---- deterministic cross-wave reduce + divide + store (first 32 threads)
    if (tid < 32) {
        const int which = tid >> 4;    // 0 = global view, 1 = local view
        const int col   = tid & 15;
        float sum = 0.0f;
        #pragma unroll
        for (int w = 0; w < WAVES; ++w) sum += part[w][which][col];

        const int len   = (which == 0) ? glen : s_slen;
        const float den = (float)(len > 0 ? len : 1);
        out[(size_t)b * (2 * H) + which * H + hbase + col] = sum / den;
    }
}

extern "C" void kernel_launch(void* const* d_in, const int* in_sizes, int n_in,
                              void* d_out, int out_size, void* d_ws, size_t ws_size,
                              hipStream_t stream) {
    (void)in_sizes; (void)n_in; (void)d_ws; (void)ws_size; (void)out_size;
    const float*         x       = (const float*)d_in[0];
    const int*           lengths = (const int*)d_in[1];
    const unsigned char* mask    = (const unsigned char*)d_in[2];
    float*               out     = (float*)d_out;

    dim3 grid(B * NTILES_H);   // 512 blocks
    dim3 block(BLOCK);         // 512 threads = 16 wave32
    GlobalLocalPool_kernel<<<grid, block, 0, stream>>>(x, lengths, mask, out);
}